// NASAdapter_51762945851485
// MI455X (gfx1250) — compile-verified
//
#include <hip/hip_runtime.h>
#include <hip/hip_bf16.h>

// ---------------------------------------------------------------------------
// GDAS NAS adapter forward for MI455X (gfx1250, wave32, WMMA).
// out = branch[argmax(gate)]( relu-or-raw x ) [+BN] + x   (hardwts is one-hot
// in value, so the scalar mixing reduces to the selected branch exactly).
// Conv branches are fp32 GEMMs -> V_WMMA_F32_16X16X4_F32 (fp32-exact).
//   - halo-widened B tile staged ONCE per K-slab, taps read as LDS shifts
//   - all k taps of the weight slab staged together (pitch 16k+1, no bank
//     conflicts) -> up to 112 WMMAs per barrier region per wave
//   - 32x32 wave tile (4 accumulators) halves LDS fragment traffic per FLOP
// ---------------------------------------------------------------------------

typedef __attribute__((ext_vector_type(2))) float v2f;
typedef __attribute__((ext_vector_type(8))) float v8f;

#define HCH   768
#define SEQ   512
#define BATCH 8

#define BM 64
#define BN 128
#define KT 16
#define BPITCH (BN + 7)          // 128 + 2*3 halo + 1 pad

// ---------------------------------------------------------------------------
// 1) Gumbel straight-through gate: index = argmax(log_softmax(arch)+gumbel(u))
// ---------------------------------------------------------------------------
__global__ void gate_kernel(const float* __restrict__ arch,
                            const float* __restrict__ u,
                            int* __restrict__ gate) {
  if (threadIdx.x != 0) return;
  float a[10];
  float mx = -__builtin_inff();
  for (int j = 0; j < 10; ++j) { a[j] = arch[j]; mx = fmaxf(mx, a[j]); }
  float se = 0.f;
  for (int j = 0; j < 10; ++j) se += expf(a[j] - mx);
  float lse = logf(se);
  int best = 0;
  float bestv = -__builtin_inff();
  for (int j = 0; j < 10; ++j) {
    float uc = fminf(fmaxf(u[j], 1e-9f), 1.f - 1e-9f);
    float g  = -logf(-logf(uc));
    float lg = (a[j] - mx - lse + g) * 0.1f;   // /TEM (softmax is monotone)
    if (lg > bestv) { bestv = lg; best = j; }  // first-occurrence argmax
  }
  *gate = best;
}

// ---------------------------------------------------------------------------
// 2) Cheap branches: none(0), avg_pool3(1), max_pool3(2), skip(9).
// ---------------------------------------------------------------------------
__global__ __launch_bounds__(256)
void simple_branch_kernel(const float* __restrict__ x,
                          const int* __restrict__ gate,
                          float* __restrict__ out) {
  int index = *gate;
  if (index >= 3 && index <= 8) return;
  size_t idx = (size_t)blockIdx.x * 256 + threadIdx.x;
  if (idx >= (size_t)BATCH * SEQ * HCH) return;
  size_t bs = idx / HCH;
  int s = (int)(bs % SEQ);
  float x0 = x[idx];
  float o;
  if (index == 0)       o = x0;          // zeros + residual
  else if (index == 9)  o = 2.f * x0;    // skip + residual
  else if (index == 1) {                 // avg_pool3, count_include_pad
    float xm = (s > 0)       ? x[idx - HCH] : 0.f;
    float xp = (s < SEQ - 1) ? x[idx + HCH] : 0.f;
    o = (xm + x0 + xp) * (1.f / 3.f) + x0;
  } else {                               // max_pool3, -inf pad
    float xm = (s > 0)       ? x[idx - HCH] : -__builtin_inff();
    float xp = (s < SEQ - 1) ? x[idx + HCH] : -__builtin_inff();
    o = fmaxf(fmaxf(xm, x0), xp) + x0;
  }
  out[idx] = o;
}

// ---------------------------------------------------------------------------
// 3) Conv branches as WMMA GEMM.  y[b,o,s] (pre-BN) -> workspace.
//    nor_conv_k : A[o,(i,t)] = Wn[o,i,t], B[(i),s'] = relu(x[b,s',i]) (halo)
//    dil_conv_k : A[o,i]     = Wp[o,i],   B[i,s] = depthwise-dilated(relu x)
// ---------------------------------------------------------------------------
__global__ __launch_bounds__(256)
void conv_gemm_kernel(const float* __restrict__ x,
                      const float* __restrict__ wn3, const float* __restrict__ wn5,
                      const float* __restrict__ wn7,
                      const float* __restrict__ wd3, const float* __restrict__ wp3,
                      const float* __restrict__ wd5, const float* __restrict__ wp5,
                      const float* __restrict__ wd7, const float* __restrict__ wp7,
                      const int* __restrict__ gate,
                      float* __restrict__ y) {
  int index = *gate;
  if (index < 3 || index > 8) return;          // uniform (scalar) exit
  const bool isDil = (index >= 6);
  const int  k     = 3 + 2 * ((index - 3) % 3);
  const float* Wn = (index == 3) ? wn3 : (index == 4) ? wn5 : wn7;
  const float* Wd = (index == 6) ? wd3 : (index == 7) ? wd5 : wd7;
  const float* Wp = (index == 6) ? wp3 : (index == 7) ? wp5 : wp7;

  const int taps = isDil ? 1 : k;
  const int pad  = isDil ? 0 : (k >> 1);
  const int AP   = KT * taps + 1;        // A row pitch: gcd(16k+1, 64)=1
  const int Wb   = BN + 2 * pad;         // staged B width (with halo)

  __shared__ float As[BM * (KT * 7 + 1)];  // all taps of one K-slab (28.9 KB)
  __shared__ float Bs[KT * BPITCH];        // halo-widened input slab (8.6 KB)

  const int tid  = threadIdx.x;
  const int lane = tid & 31;
  const int wave = tid >> 5;     // 8 waves
  const int mw   = wave >> 2;    // 0..1 : 32-row slice of M-tile
  const int nw   = wave & 3;     // 0..3 : 32-col slice of N-tile

  const int b     = blockIdx.z;
  const int oBase = blockIdx.x * BM;
  const int sBase = blockIdx.y * BN;

  v8f acc00 = {}, acc01 = {}, acc10 = {}, acc11 = {};

  const int mA = lane & 15;            // fragment row (A) / col (B)
  const int kh = (lane >> 4) << 1;     // lanes 16..31 hold K=2,3

  for (int i0 = 0; i0 < HCH; i0 += KT) {
    // ---- stage A slab (all taps): As[m*AP + kk*taps + t] ----
    for (int e = tid; e < BM * KT * taps; e += 256) {
      int rem = e % (KT * taps);       // = kk*taps + t  (t fastest -> coalesced)
      int m   = e / (KT * taps);
      float av = isDil ? Wp[(oBase + m) * HCH + (i0 + rem)]
                       : Wn[((size_t)(oBase + m) * HCH + i0) * k + rem];
      As[m * AP + rem] = av;
    }
    // ---- stage halo-widened B slab: Bs[kk*BPITCH + c] ----
    for (int e = tid; e < KT * Wb; e += 256) {
      int kk = e & 15, c = e >> 4;     // kk fastest -> coalesced x reads
      int i = i0 + kk;
      float bv = 0.f;
      if (!isDil) {
        int s = sBase + c - pad;
        if (s >= 0 && s < SEQ) {
          size_t gi = ((size_t)b * SEQ + s) * HCH + i;
          bv = fmaxf(x[gi], 0.f);
          if (i0 + KT < HCH) __builtin_prefetch(&x[gi + KT], 0, 0);
        }
      } else {
        // fused depthwise dilated conv (dilation 2, pad k-1)
        for (int tt = 0; tt < k; ++tt) {
          int s = sBase + c + 2 * tt - (k - 1);
          if (s >= 0 && s < SEQ) {
            float xv = fmaxf(x[((size_t)b * SEQ + s) * HCH + i], 0.f);
            bv += Wd[i * k + tt] * xv;
          }
        }
      }
      Bs[kk * BPITCH + c] = bv;
    }
    __syncthreads();
    // ---- taps x 4 K-steps x 4 WMMA per wave per barrier region ----
    for (int t = 0; t < taps; ++t) {
      const int cb = nw * 32 + mA + t;           // tap-shifted halo column
#pragma unroll
      for (int kk = 0; kk < KT; kk += 4) {
        int ka = (kk + kh) * taps + t;
        v2f a0, a1, b0, b1;
        a0.x = As[(mw * 32 + mA) * AP + ka];
        a0.y = As[(mw * 32 + mA) * AP + ka + taps];
        a1.x = As[(mw * 32 + 16 + mA) * AP + ka];
        a1.y = As[(mw * 32 + 16 + mA) * AP + ka + taps];
        b0.x = Bs[(kk + kh) * BPITCH + cb];
        b0.y = Bs[(kk + kh + 1) * BPITCH + cb];
        b1.x = Bs[(kk + kh) * BPITCH + cb + 16];
        b1.y = Bs[(kk + kh + 1) * BPITCH + cb + 16];
        acc00 = __builtin_amdgcn_wmma_f32_16x16x4_f32(
            false, a0, false, b0, (short)0, acc00, false, false);
        acc01 = __builtin_amdgcn_wmma_f32_16x16x4_f32(
            false, a0, false, b1, (short)0, acc01, false, false);
        acc10 = __builtin_amdgcn_wmma_f32_16x16x4_f32(
            false, a1, false, b0, (short)0, acc10, false, false);
        acc11 = __builtin_amdgcn_wmma_f32_16x16x4_f32(
            false, a1, false, b1, (short)0, acc11, false, false);
      }
    }
    __syncthreads();
  }

  // ---- store C/D: VGPR j -> M = j + 8*(lane>>4), N = lane&15 ----
  const int nlo = lane & 15;
  const int mhi = (lane >> 4) * 8;
#pragma unroll
  for (int j = 0; j < 8; ++j) {
    int o0 = oBase + mw * 32 + mhi + j;
    int s0 = sBase + nw * 32 + nlo;
    size_t base0 = ((size_t)b * HCH + o0) * SEQ + s0;
    size_t base1 = ((size_t)b * HCH + o0 + 16) * SEQ + s0;
    y[base0]      = acc00[j];
    y[base0 + 16] = acc01[j];
    y[base1]      = acc10[j];
    y[base1 + 16] = acc11[j];
  }
}

// ---------------------------------------------------------------------------
// 4) Training-mode BN stats per channel over (B, S): mean + rsqrt(var+eps)
// ---------------------------------------------------------------------------
__global__ __launch_bounds__(256)
void bn_stats_kernel(const float* __restrict__ y, const int* __restrict__ gate,
                     float* __restrict__ mean, float* __restrict__ rsig) {
  int index = *gate;
  if (index < 3 || index > 8) return;
  int c = blockIdx.x;
  __shared__ float ssum[256], ssq[256];
  float s = 0.f, q = 0.f;
  for (int e = threadIdx.x; e < BATCH * SEQ; e += 256) {
    int b = e / SEQ, p = e % SEQ;
    float v = y[((size_t)b * HCH + c) * SEQ + p];
    s += v; q += v * v;
  }
  ssum[threadIdx.x] = s; ssq[threadIdx.x] = q;
  __syncthreads();
  for (int st = 128; st > 0; st >>= 1) {
    if (threadIdx.x < st) {
      ssum[threadIdx.x] += ssum[threadIdx.x + st];
      ssq[threadIdx.x]  += ssq[threadIdx.x + st];
    }
    __syncthreads();
  }
  if (threadIdx.x == 0) {
    const float inv = 1.f / (float)(BATCH * SEQ);
    float m = ssum[0] * inv;
    float v = ssq[0] * inv - m * m;          // population var (ddof=0)
    mean[c] = m;
    rsig[c] = rsqrtf(v + 1e-5f);
  }
}

// ---------------------------------------------------------------------------
// 5) Normalize + transpose back + residual
// ---------------------------------------------------------------------------
__global__ __launch_bounds__(256)
void bn_finalize_kernel(const float* __restrict__ x, const float* __restrict__ y,
                        const int* __restrict__ gate,
                        const float* g3, const float* b3, const float* g5,
                        const float* b5, const float* g7, const float* b7,
                        const float* gd3, const float* bd3, const float* gd5,
                        const float* bd5, const float* gd7, const float* bd7,
                        const float* __restrict__ mean,
                        const float* __restrict__ rsig,
                        float* __restrict__ out) {
  int index = *gate;
  if (index < 3 || index > 8) return;
  const float* G;
  const float* Bt;
  switch (index) {
    case 3: G = g3;  Bt = b3;  break;
    case 4: G = g5;  Bt = b5;  break;
    case 5: G = g7;  Bt = b7;  break;
    case 6: G = gd3; Bt = bd3; break;
    case 7: G = gd5; Bt = bd5; break;
    default: G = gd7; Bt = bd7; break;
  }
  size_t idx = (size_t)blockIdx.x * 256 + threadIdx.x;
  if (idx >= (size_t)BATCH * SEQ * HCH) return;
  int h = (int)(idx % HCH);
  size_t bs = idx / HCH;
  int s = (int)(bs % SEQ);
  int b = (int)(bs / SEQ);
  float yv = y[((size_t)b * HCH + h) * SEQ + s];
  out[idx] = (yv - mean[h]) * rsig[h] * G[h] + Bt[h] + x[idx];
}

// ---------------------------------------------------------------------------
extern "C" void kernel_launch(void* const* d_in, const int* in_sizes, int n_in,
                              void* d_out, int out_size, void* d_ws, size_t ws_size,
                              hipStream_t stream) {
  const float* x    = (const float*)d_in[0];
  const float* u    = (const float*)d_in[1];
  const float* arch = (const float*)d_in[2];
  const float* wn3 = (const float*)d_in[3];
  const float* g3  = (const float*)d_in[4];
  const float* b3  = (const float*)d_in[5];
  const float* wn5 = (const float*)d_in[6];
  const float* g5  = (const float*)d_in[7];
  const float* b5  = (const float*)d_in[8];
  const float* wn7 = (const float*)d_in[9];
  const float* g7  = (const float*)d_in[10];
  const float* b7  = (const float*)d_in[11];
  const float* wd3 = (const float*)d_in[12];
  const float* wp3 = (const float*)d_in[13];
  const float* gd3 = (const float*)d_in[14];
  const float* bd3 = (const float*)d_in[15];
  const float* wd5 = (const float*)d_in[16];
  const float* wp5 = (const float*)d_in[17];
  const float* gd5 = (const float*)d_in[18];
  const float* bd5 = (const float*)d_in[19];
  const float* wd7 = (const float*)d_in[20];
  const float* wp7 = (const float*)d_in[21];
  const float* gd7 = (const float*)d_in[22];
  const float* bd7 = (const float*)d_in[23];
  float* out = (float*)d_out;

  // workspace layout
  int*   gate = (int*)d_ws;
  float* mean = (float*)((char*)d_ws + 256);
  float* rsig = mean + HCH;
  float* y    = (float*)((char*)d_ws + 8192);   // [B][H][S] fp32, ~12.6 MB

  const size_t N = (size_t)BATCH * SEQ * HCH;
  const int nb = (int)((N + 255) / 256);

  gate_kernel<<<1, 32, 0, stream>>>(arch, u, gate);
  simple_branch_kernel<<<nb, 256, 0, stream>>>(x, gate, out);
  conv_gemm_kernel<<<dim3(HCH / BM, SEQ / BN, BATCH), 256, 0, stream>>>(
      x, wn3, wn5, wn7, wd3, wp3, wd5, wp5, wd7, wp7, gate, y);
  bn_stats_kernel<<<HCH, 256, 0, stream>>>(y, gate, mean, rsig);
  bn_finalize_kernel<<<nb, 256, 0, stream>>>(
      x, y, gate, g3, b3, g5, b5, g7, b7, gd3, bd3, gd5, bd5, gd7, bd7,
      mean, rsig, out);
}